// GraphAttentionNetwork_23106924053199
// MI455X (gfx1250) — compile-verified
//
#include <hip/hip_runtime.h>

// ---------------- problem constants (from reference) ----------------
#define Nn   20000      // nodes
#define Ne   320000     // edges
#define FIN  128        // input feature dim
#define HIDC 64         // layer1 per-head channels
#define FOUT 128        // layer2 per-head channels
#define NH   3          // heads
#define IN2  320        // FIN + NH*HIDC
#define HC1  (NH*HIDC)  // 192
#define HC2  (NH*FOUT)  // 384
#define NEGS 0.15f      // LeakyReLU slope

typedef __attribute__((ext_vector_type(2))) float v2f;
typedef __attribute__((ext_vector_type(8))) float v8f;

// ---------------- helpers ----------------
__device__ __forceinline__ unsigned fkey(float f) {
    unsigned u = __float_as_uint(f);
    return (u & 0x80000000u) ? ~u : (u | 0x80000000u);   // order-preserving map
}
__device__ __forceinline__ float fdec(unsigned k) {
    return (k & 0x80000000u) ? __uint_as_float(k & 0x7fffffffu)
                             : __uint_as_float(~k);
}
#define KEY_NEGINF 0x007FFFFFu   // fkey(-inf)

__global__ void fill_f32(float* __restrict__ p, int n, float v) {
    int i = blockIdx.x * blockDim.x + threadIdx.x;
    if (i < n) p[i] = v;
}
__global__ void fill_u32(unsigned* __restrict__ p, int n, unsigned v) {
    int i = blockIdx.x * blockDim.x + threadIdx.x;
    if (i < n) p[i] = v;
}

// ---------------- f32 WMMA GEMM:  D[M,Nc] = A[M,K] @ B[K,Nc] + bias[Nc] ----
// 5 waves / block, one 16x16 D-tile per wave, B panel staged in LDS.
// Requires M%80==0, K%4==0, Nc%16==0 (true here: 20000, 128/320, 192/384).
#define GEMM_WAVES 5
__global__ __launch_bounds__(32 * GEMM_WAVES)
void gemm_wmma_f32(const float* __restrict__ A, const float* __restrict__ B,
                   const float* __restrict__ bias, float* __restrict__ D,
                   int K, int Nc) {
    extern __shared__ float ldsB[];               // K*16 floats
    const int tid  = threadIdx.x;
    const int wave = tid >> 5;
    const int lane = tid & 31;
    const int lo   = lane & 15;                   // N (and A-row) sub-index
    const int hi   = lane >> 4;                   // K-pair selector
    const int colBase = blockIdx.y * 16;

    // stage B panel: B[k][colBase+n] -> ldsB[k*16+n]  (coalesced over n)
    for (int i = tid; i < K * 16; i += blockDim.x) {
        int k = i >> 4, n = i & 15;
        ldsB[i] = B[(size_t)k * Nc + colBase + n];
    }
    __syncthreads();

    const int row0 = (blockIdx.x * GEMM_WAVES + wave) * 16;
    const float* Arow = A + (size_t)(row0 + lo) * K;   // A-frag: M = lane&15

    v8f acc = {0.f, 0.f, 0.f, 0.f, 0.f, 0.f, 0.f, 0.f};
    for (int k0 = 0; k0 < K; k0 += 4) {
        const int ka = k0 + (hi << 1);            // ISA: V0 holds K=0(lo)/K=2(hi)
        __builtin_prefetch(Arow + ka + 16, 0, 3); // speculative A prefetch
        v2f a; a.x = Arow[ka];            a.y = Arow[ka + 1];
        v2f b; b.x = ldsB[ka * 16 + lo];  b.y = ldsB[(ka + 1) * 16 + lo];
        acc = __builtin_amdgcn_wmma_f32_16x16x4_f32(
                  false, a, false, b, (short)0, acc, false, false);
    }

    const float bv = bias[colBase + lo];
    float* Dp = D + (size_t)(row0 + (hi << 3)) * Nc + colBase + lo; // M=r+8*hi
#pragma unroll
    for (int r = 0; r < 8; ++r) Dp[(size_t)r * Nc] = acc[r] + bv;
}

// ---------------- edge pass 1: logits + segment max ----------------
__global__ void edge_logits(const float* __restrict__ xl, const float* __restrict__ xr,
                            const int* __restrict__ src, const int* __restrict__ dst,
                            const float* __restrict__ att,
                            float* __restrict__ logits, unsigned* __restrict__ maxkey,
                            int C) {
    int t = blockIdx.x * blockDim.x + threadIdx.x;
    if (t >= Ne * NH) return;
    const int e = t / NH, h = t - e * NH;
    const int s = src[e], d = dst[e];
    const int HC = NH * C;
    const float* pl = xl + (size_t)s * HC + h * C;
    const float* pr = xr + (size_t)d * HC + h * C;
    const float* pa = att + h * C;
    float acc = 0.f;
#pragma unroll 4
    for (int c = 0; c < C; c += 4) {
        float4 a = *(const float4*)(pl + c);
        float4 b = *(const float4*)(pr + c);
        float4 w = *(const float4*)(pa + c);
        float e0 = a.x + b.x; e0 = (e0 > 0.f) ? e0 : NEGS * e0; acc += e0 * w.x;
        float e1 = a.y + b.y; e1 = (e1 > 0.f) ? e1 : NEGS * e1; acc += e1 * w.y;
        float e2 = a.z + b.z; e2 = (e2 > 0.f) ? e2 : NEGS * e2; acc += e2 * w.z;
        float e3 = a.w + b.w; e3 = (e3 > 0.f) ? e3 : NEGS * e3; acc += e3 * w.w;
    }
    logits[t] = acc;
    atomicMax(maxkey + (size_t)d * NH + h, fkey(acc));
}

// ---------------- edge pass 2: exp + segment sum ----------------
__global__ void edge_expsum(float* __restrict__ logits, const int* __restrict__ dst,
                            const unsigned* __restrict__ maxkey, float* __restrict__ denom) {
    int t = blockIdx.x * blockDim.x + threadIdx.x;
    if (t >= Ne * NH) return;
    const int e = t / NH, h = t - e * NH;
    const int d = dst[e];
    const float m  = fdec(maxkey[(size_t)d * NH + h]);
    const float ex = __expf(logits[t] - m);
    logits[t] = ex;                                   // reuse buffer for ex
    atomicAdd(denom + (size_t)d * NH + h, ex);
}

// ---------------- edge pass 3: alpha-weighted scatter ----------------
__global__ void edge_scatter(const float* __restrict__ xl,
                             const int* __restrict__ src, const int* __restrict__ dst,
                             const float* __restrict__ exbuf, const float* __restrict__ denom,
                             float* __restrict__ acc, int C) {
    int t = blockIdx.x * blockDim.x + threadIdx.x;
    if (t >= Ne * NH) return;
    const int e = t / NH, h = t - e * NH;
    const int s = src[e], d = dst[e];
    const int HC = NH * C;
    const float alpha = exbuf[t] / (denom[(size_t)d * NH + h] + 1e-16f);
    const float* pl = xl + (size_t)s * HC + h * C;
    float* pd = acc + (size_t)d * HC + h * C;
#pragma unroll 4
    for (int c = 0; c < C; c += 4) {
        float4 v = *(const float4*)(pl + c);
        atomicAdd(pd + c + 0, alpha * v.x);
        atomicAdd(pd + c + 1, alpha * v.y);
        atomicAdd(pd + c + 2, alpha * v.z);
        atomicAdd(pd + c + 3, alpha * v.w);
    }
}

// ---------------- h = [x, relu(gat1 + bias1)] ----------------
__global__ void concat_relu(const float* __restrict__ x, const float* __restrict__ acc1,
                            const float* __restrict__ bias1, float* __restrict__ hcat) {
    int t = blockIdx.x * blockDim.x + threadIdx.x;
    if (t >= Nn * IN2) return;
    const int n = t / IN2, j = t - n * IN2;
    float v;
    if (j < FIN) v = x[(size_t)n * FIN + j];
    else {
        const int k = j - FIN;
        v = fmaxf(acc1[(size_t)n * HC1 + k] + bias1[k], 0.f);
    }
    hcat[t] = v;
}

// ---------------- out = relu(mean_h(gat2) + bias2) ----------------
__global__ void head_mean_relu(const float* __restrict__ acc2,
                               const float* __restrict__ bias2, float* __restrict__ out) {
    int t = blockIdx.x * blockDim.x + threadIdx.x;
    if (t >= Nn * FOUT) return;
    const int n = t / FOUT, c = t - n * FOUT;
    const float* p = acc2 + (size_t)n * HC2;
    float v = (p[c] + p[FOUT + c] + p[2 * FOUT + c]) * (1.f / 3.f) + bias2[c];
    out[t] = fmaxf(v, 0.f);
}

// ---------------- launch ----------------
extern "C" void kernel_launch(void* const* d_in, const int* in_sizes, int n_in,
                              void* d_out, int out_size, void* d_ws, size_t ws_size,
                              hipStream_t stream) {
    const float* x     = (const float*)d_in[0];
    const float* Wl1   = (const float*)d_in[1];
    const float* bl1   = (const float*)d_in[2];
    const float* Wr1   = (const float*)d_in[3];
    const float* br1   = (const float*)d_in[4];
    const float* att1  = (const float*)d_in[5];
    const float* bias1 = (const float*)d_in[6];
    const float* Wl2   = (const float*)d_in[7];
    const float* bl2   = (const float*)d_in[8];
    const float* Wr2   = (const float*)d_in[9];
    const float* br2   = (const float*)d_in[10];
    const float* att2  = (const float*)d_in[11];
    const float* bias2 = (const float*)d_in[12];  // see note: dict order below
    const int*   ei    = (const int*)d_in[12];

    // setup_inputs order: 0 x, 1 Wl1, 2 bl1, 3 Wr1, 4 br1, 5 att1, 6 bias1,
    //                     7 Wl2, 8 bl2, 9 Wr2, 10 br2, 11 att2, 12 bias2, 13 edge_index
    bias2 = (const float*)d_in[12];
    const int* eidx = (const int*)d_in[13];
    const int* src = eidx;            // edge_index[0]
    const int* dst = eidx + Ne;       // edge_index[1]
    (void)ei; (void)in_sizes; (void)n_in; (void)out_size; (void)ws_size;

    // workspace layout (floats, every region 64B aligned)
    float* ws   = (float*)d_ws;
    float* XL1  = ws;
    float* XR1  = XL1  + (size_t)Nn * HC1;
    float* ACC1 = XR1  + (size_t)Nn * HC1;
    float* HCAT = ACC1 + (size_t)Nn * HC1;
    float* XL2  = HCAT + (size_t)Nn * IN2;
    float* XR2  = XL2  + (size_t)Nn * HC2;
    float* ACC2 = XR2  + (size_t)Nn * HC2;
    float* LOGB = ACC2 + (size_t)Nn * HC2;       // Ne*NH (logits, then ex)
    float* DEN  = LOGB + (size_t)Ne * NH;        // Nn*NH
    unsigned* MAXK = (unsigned*)(DEN + (size_t)Nn * NH);

    const int EH  = Ne * NH;
    const int NHn = Nn * NH;
    dim3 blk(256);
    auto g = [](int n) { return dim3((unsigned)((n + 255) / 256)); };

    // ---------------- layer 1 ----------------
    fill_f32<<<g(Nn * HC1), blk, 0, stream>>>(ACC1, Nn * HC1, 0.f);
    fill_f32<<<g(NHn), blk, 0, stream>>>(DEN, NHn, 0.f);
    fill_u32<<<g(NHn), blk, 0, stream>>>(MAXK, NHn, KEY_NEGINF);

    dim3 gg1(Nn / (16 * GEMM_WAVES), HC1 / 16);
    size_t lds1 = (size_t)FIN * 16 * sizeof(float);
    gemm_wmma_f32<<<gg1, 32 * GEMM_WAVES, lds1, stream>>>(x, Wl1, bl1, XL1, FIN, HC1);
    gemm_wmma_f32<<<gg1, 32 * GEMM_WAVES, lds1, stream>>>(x, Wr1, br1, XR1, FIN, HC1);

    edge_logits <<<g(EH), blk, 0, stream>>>(XL1, XR1, src, dst, att1, LOGB, MAXK, HIDC);
    edge_expsum <<<g(EH), blk, 0, stream>>>(LOGB, dst, MAXK, DEN);
    edge_scatter<<<g(EH), blk, 0, stream>>>(XL1, src, dst, LOGB, DEN, ACC1, HIDC);

    concat_relu<<<g(Nn * IN2), blk, 0, stream>>>(x, ACC1, bias1, HCAT);

    // ---------------- layer 2 ----------------
    fill_f32<<<g(Nn * HC2), blk, 0, stream>>>(ACC2, Nn * HC2, 0.f);
    fill_f32<<<g(NHn), blk, 0, stream>>>(DEN, NHn, 0.f);
    fill_u32<<<g(NHn), blk, 0, stream>>>(MAXK, NHn, KEY_NEGINF);

    dim3 gg2(Nn / (16 * GEMM_WAVES), HC2 / 16);
    size_t lds2 = (size_t)IN2 * 16 * sizeof(float);
    gemm_wmma_f32<<<gg2, 32 * GEMM_WAVES, lds2, stream>>>(HCAT, Wl2, bl2, XL2, IN2, HC2);
    gemm_wmma_f32<<<gg2, 32 * GEMM_WAVES, lds2, stream>>>(HCAT, Wr2, br2, XR2, IN2, HC2);

    edge_logits <<<g(EH), blk, 0, stream>>>(XL2, XR2, src, dst, att2, LOGB, MAXK, FOUT);
    edge_expsum <<<g(EH), blk, 0, stream>>>(LOGB, dst, MAXK, DEN);
    edge_scatter<<<g(EH), blk, 0, stream>>>(XL2, src, dst, LOGB, DEN, ACC2, FOUT);

    head_mean_relu<<<g(Nn * FOUT), blk, 0, stream>>>(ACC2, bias2, (float*)d_out);
}